// VolumeRenderer_8735963480354
// MI455X (gfx1250) — compile-verified
//
#include <hip/hip_runtime.h>

// -----------------------------------------------------------------------------
// NeRF volume rendering for MI455X (gfx1250, wave32).
//
// Shapes: B=8, N=8192, S=128  -> 65536 rays, 128 samples each.
// Mapping: one wave32 per ray; lane L owns 4 consecutive samples [4L, 4L+3].
//   - sigma / t_vals: one float4 (b128) coalesced load per lane
//   - rgbs: staged into LDS via CDNA5 GLOBAL_LOAD_ASYNC_TO_LDS_B128 (ASYNCcnt)
//           when the builtin is available; falls back to b128 load + ds_store.
//           (lane L consumes the 12 floats [12L, 12L+12) from LDS)
//   - exclusive cumprod: lane-local prefix over 4 samples + 5-step wave32
//     multiplicative Kogge-Stone scan via __shfl_up (LDS permute HW)
//   - rgb/depth/acc: 5-step __shfl_xor butterfly reductions, lane 0 writes
//   - weights: one non-temporal float4 (b128) coalesced store per lane
//     (written once, never re-read -> keep it out of L2)
//
// Roofline: ~195 MB moved, ~0.1 GFLOP (+8.4M exp) -> pure HBM-bandwidth bound
// (~8.4 us at 23.3 TB/s). No matmul structure exists (per-ray batched dot
// products, no shared operand), so WMMA is intentionally not used.
// -----------------------------------------------------------------------------

#define VR_BN 65536              // B*N
#define VR_S  128                // samples per ray
#define VR_WAVES_PER_BLOCK 8     // 256 threads = 8 wave32
#define VR_BLOCK (VR_WAVES_PER_BLOCK * 32)

#if defined(__HIP_DEVICE_COMPILE__) && \
    __has_builtin(__builtin_amdgcn_global_load_async_to_lds_b128)
#define VR_ASYNC_LDS 1
#else
#define VR_ASYNC_LDS 0
#endif

typedef float vr_v4f __attribute__((ext_vector_type(4)));
typedef int   vr_v4i __attribute__((ext_vector_type(4)));

__global__ __launch_bounds__(VR_BLOCK, 2)
void VolumeRenderer_8735963480354_kernel(const float* __restrict__ densities,
                                         const float* __restrict__ rgbs,
                                         const float* __restrict__ t_vals,
                                         float* __restrict__ out_rgb,    // [BN,3]
                                         float* __restrict__ out_w,      // [BN,S]
                                         float* __restrict__ out_depth,  // [BN]
                                         float* __restrict__ out_acc)    // [BN]
{
    __shared__ float lds_rgb[VR_WAVES_PER_BLOCK][VR_S * 3];

    const int lane = threadIdx.x & 31;
    const int wave = threadIdx.x >> 5;
    const int ray  = blockIdx.x * VR_WAVES_PER_BLOCK + wave;

    // ---- coalesced 128-bit global loads -------------------------------------
    const float4* sig4 = (const float4*)(densities + (size_t)ray * VR_S);
    const float4* t4p  = (const float4*)(t_vals    + (size_t)ray * VR_S);
    const float4* r4p  = (const float4*)(rgbs      + (size_t)ray * VR_S * 3);

    // ---- rgb staging into LDS (96 float4 per ray) ---------------------------
    // Global and LDS strides between the 3 transfers are both 512B, and the
    // async instruction adds its immediate offset to BOTH addresses, so one
    // address setup serves all three transfers.
#if VR_ASYNC_LDS
    {
        typedef __attribute__((address_space(1))) vr_v4i gv4i;  // global (AS1)
        typedef __attribute__((address_space(3))) vr_v4i lv4i;  // LDS    (AS3)
        gv4i* gsrc = (gv4i*)((float4*)(r4p + lane));
        lv4i* ldst = (lv4i*)(((float4*)&lds_rgb[wave][0]) + lane);
        __builtin_amdgcn_global_load_async_to_lds_b128(gsrc, ldst, 0,    0);
        __builtin_amdgcn_global_load_async_to_lds_b128(gsrc, ldst, 512,  0);
        __builtin_amdgcn_global_load_async_to_lds_b128(gsrc, ldst, 1024, 0);
    }
#else
    {
        float4 r0 = r4p[lane];
        float4 r1 = r4p[lane + 32];
        float4 r2 = r4p[lane + 64];
        float4* ldst = (float4*)&lds_rgb[wave][0];
        ldst[lane]      = r0;
        ldst[lane + 32] = r1;
        ldst[lane + 64] = r2;
    }
#endif

    float4 sg = sig4[lane];           // sigma for samples 4L..4L+3
    float4 tt = t4p[lane];            // t     for samples 4L..4L+3

    // ---- deltas (t[i+1]-t[i], last = 1e10) ----------------------------------
    const float INF_DELTA = 1e10f;
    float tnext = __shfl_down(tt.x, 1, 32);   // lane L+1's first t
    float d0 = tt.y - tt.x;
    float d1 = tt.z - tt.y;
    float d2 = tt.w - tt.z;
    float d3 = (lane == 31) ? INF_DELTA : (tnext - tt.w);

    // ---- alpha and survival factors -----------------------------------------
    const float EPS = 1e-10f;
    float e0 = __expf(-sg.x * d0);    // 1 - alpha
    float e1 = __expf(-sg.y * d1);
    float e2 = __expf(-sg.z * d2);
    float e3 = __expf(-sg.w * d3);
    float a0 = 1.0f - e0, a1 = 1.0f - e1, a2 = 1.0f - e2, a3 = 1.0f - e3;
    float u0 = e0 + EPS,  u1 = e1 + EPS,  u2 = e2 + EPS,  u3 = e3 + EPS;

    // lane-local prefix products over the 4 owned samples
    float l1 = u0;
    float l2 = l1 * u1;
    float l3 = l2 * u2;
    float ltot = l3 * u3;

    // ---- wave32 multiplicative Kogge-Stone scan of lane totals --------------
    float x = ltot;
    #pragma unroll
    for (int off = 1; off < 32; off <<= 1) {
        float y = __shfl_up(x, off, 32);
        if (lane >= off) x *= y;
    }
    float excl = __shfl_up(x, 1, 32);          // exclusive product across lanes
    if (lane == 0) excl = 1.0f;

    // transmittance and weights for the 4 samples
    float T0 = excl;
    float T1 = excl * l1;
    float T2 = excl * l2;
    float T3 = excl * l3;
    float w0 = T0 * a0, w1 = T1 * a1, w2 = T2 * a2, w3 = T3 * a3;

    // ---- coalesced 128-bit non-temporal weights store -----------------------
    {
        vr_v4f wv = {w0, w1, w2, w3};
        vr_v4f* wout = (vr_v4f*)(out_w + (size_t)ray * VR_S);
        __builtin_nontemporal_store(wv, wout + lane);
    }

    // ---- wait for async rgb staging, make it visible across lanes -----------
#if VR_ASYNC_LDS
#if __has_builtin(__builtin_amdgcn_s_wait_asynccnt)
    __builtin_amdgcn_s_wait_asynccnt(0);
#else
    asm volatile("s_wait_asynccnt 0" ::: "memory");
#endif
#endif
    __syncthreads();

    const float* rp = &lds_rgb[wave][12 * lane];
    float cr  = w0 * rp[0] + w1 * rp[3] + w2 * rp[6] + w3 * rp[9];
    float cg  = w0 * rp[1] + w1 * rp[4] + w2 * rp[7] + w3 * rp[10];
    float cb  = w0 * rp[2] + w1 * rp[5] + w2 * rp[8] + w3 * rp[11];
    float dep = w0 * tt.x + w1 * tt.y + w2 * tt.z + w3 * tt.w;
    float acc = w0 + w1 + w2 + w3;

    // ---- wave32 butterfly reductions ----------------------------------------
    #pragma unroll
    for (int off = 16; off > 0; off >>= 1) {
        cr  += __shfl_xor(cr,  off, 32);
        cg  += __shfl_xor(cg,  off, 32);
        cb  += __shfl_xor(cb,  off, 32);
        dep += __shfl_xor(dep, off, 32);
        acc += __shfl_xor(acc, off, 32);
    }

    if (lane == 0) {
        out_rgb[(size_t)ray * 3 + 0] = cr;
        out_rgb[(size_t)ray * 3 + 1] = cg;
        out_rgb[(size_t)ray * 3 + 2] = cb;
        out_depth[ray] = dep;
        out_acc[ray]   = acc;
    }
}

extern "C" void kernel_launch(void* const* d_in, const int* in_sizes, int n_in,
                              void* d_out, int out_size, void* d_ws, size_t ws_size,
                              hipStream_t stream) {
    (void)in_sizes; (void)n_in; (void)out_size; (void)d_ws; (void)ws_size;

    const float* densities = (const float*)d_in[0];  // [B,N,S,1]
    const float* rgbs      = (const float*)d_in[1];  // [B,N,S,3]
    const float* t_vals    = (const float*)d_in[2];  // [B,N,S]

    // outputs concatenated flat in return order: rgb, weights, depth, acc
    float* out       = (float*)d_out;
    float* out_rgb   = out;                                   // BN*3
    float* out_w     = out_rgb + (size_t)VR_BN * 3;           // BN*S
    float* out_depth = out_w   + (size_t)VR_BN * VR_S;        // BN
    float* out_acc   = out_depth + (size_t)VR_BN;             // BN

    dim3 grid(VR_BN / VR_WAVES_PER_BLOCK);   // 8192 blocks, exact
    dim3 block(VR_BLOCK);                    // 256 threads = 8 wave32
    VolumeRenderer_8735963480354_kernel<<<grid, block, 0, stream>>>(
        densities, rgbs, t_vals, out_rgb, out_w, out_depth, out_acc);
}